// signature_model_2542620639655
// MI455X (gfx1250) — compile-verified
//
#include <hip/hip_runtime.h>
#include <stdint.h>

typedef __attribute__((ext_vector_type(2))) float v2f;
typedef __attribute__((ext_vector_type(8))) float v8f;
typedef __attribute__((ext_vector_type(4))) unsigned int v4u;
typedef __attribute__((ext_vector_type(8))) int v8i;
typedef __attribute__((ext_vector_type(4))) int v4i;

#define BATCH 128
#define LEN   512
#define CH    8
#define SIGD  4680   // 8 + 64 + 512 + 4096
#define NOUT  10

#if defined(__has_builtin)
#if __has_builtin(__builtin_amdgcn_tensor_load_to_lds)
#define HAVE_TDM 1
#endif
#endif

// ---------------------------------------------------------------------------
// Kernel 1: depth-4 signature of one CHUNK of one batch's path.
// grid = BATCH * nchunk blocks, 256 threads. Horner form of S (x) exp(v):
//   n4 = s4 + (s3 + (s2 + (s1 + v/4)(x)v/3)(x)v/2)(x)v
//   n3 = s3 + (s2 + (s1 + v/3)(x)v/2)(x)v ;  n2 = s2 + (s1 + v/2)(x)v
// s1,v replicated in registers; s2,s3 in LDS; s4 is a write-only accumulator
// held in registers (thread owns rows p=2t,2t+1 of the 512x8 level-4 tile).
// Level-2 scratch is double-buffered -> exactly ONE barrier per step.
// The chunk's slab of X is DMA'd into LDS by the Tensor Data Mover.
// ---------------------------------------------------------------------------
__global__ __launch_bounds__(256) void sig_chunk_kernel(const float* __restrict__ X,
                                                        float* __restrict__ csig,
                                                        int nchunk, int clen)
{
  __shared__ float Xs[LEN * CH];       // worst case (nchunk=1): 16 KB
  __shared__ float s2[64];
  __shared__ float s3[512];
  __shared__ float B4s[2][64];
  __shared__ float B3s[2][64];
  const int tid   = threadIdx.x;
  const int batch = (int)blockIdx.x / nchunk;
  const int chunk = (int)blockIdx.x % nchunk;
  const int t0 = chunk * clen;
  const int t1 = min(t0 + clen, LEN - 1);
  const int steps = t1 - t0;
  const unsigned nelem = (unsigned)(steps + 1) * CH;   // rows t0..t1 inclusive
  const float* Xb = X + (size_t)batch * (LEN * CH) + (size_t)t0 * CH;

#ifdef HAVE_TDM
  if (tid < 32) {  // wave 0 only: single TDM DMA of this chunk's slab
    unsigned long long ga = (unsigned long long)(uintptr_t)Xb;
    unsigned int lds = (unsigned int)(uintptr_t)&Xs[0];
    v4u g0 = { 1u,                                   // count=1 valid descriptor
               lds,                                  // lds_addr
               (unsigned int)(ga & 0xFFFFFFFFull),   // global_addr[31:0]
               (unsigned int)((ga >> 32) & 0x1FFFFFFull) | 0x80000000u }; // type=2
    v8i g1 = { (int)0x00020000,                 // data_size=2 (4B)
               (int)((nelem & 0xFFFFu) << 16),  // tensor_dim0 = nelem (low16)
               (int)(1u << 16),                 // dim0 hi = 0, tensor_dim1 = 1
               (int)(nelem << 16),              // tile_dim0 = nelem
               0,                               // tile_dim1/2 unused
               (int)nelem,                      // tensor_dim0_stride low 32
               0, 0 };
    v4i gz = { 0, 0, 0, 0 };
#if __clang_major__ >= 23
    v8i gz8 = { 0, 0, 0, 0, 0, 0, 0, 0 };
    __builtin_amdgcn_tensor_load_to_lds(g0, g1, gz, gz, gz8, 0);
#else
    __builtin_amdgcn_tensor_load_to_lds(g0, g1, gz, gz, 0);
#endif
    __builtin_amdgcn_s_wait_tensorcnt(0);
  }
#else
  for (unsigned i = tid; i < nelem; i += 256) Xs[i] = Xb[i];
#endif

  if (tid < 64) s2[tid] = 0.f;
  s3[2 * tid] = 0.f;
  s3[2 * tid + 1] = 0.f;
  float s1[CH];
#pragma unroll
  for (int c = 0; c < CH; ++c) s1[c] = 0.f;
  float acc4[16];
#pragma unroll
  for (int r = 0; r < 16; ++r) acc4[r] = 0.f;
  __syncthreads();

  const int p0 = 2 * tid;              // owned level-3 rows (same pair all steps)
  const int q0 = p0 >> 3;
  const int j0 = p0 & 7;               // in {0,2,4,6}

  for (int t = 0; t < steps; ++t) {
    const int buf = t & 1;
    // Phase 1 (registers, replicated): increment + Horner heads
    float v[CH], A4[CH], A3[CH], A2[CH];
#pragma unroll
    for (int c = 0; c < CH; ++c) {
      float d = Xs[(t + 1) * CH + c] - Xs[t * CH + c];
      v[c]  = d;
      A4[c] = s1[c] + 0.25f * d;
      A3[c] = s1[c] + (1.f / 3.f) * d;
      A2[c] = s1[c] + 0.5f * d;
    }
    // Phase 2: level-2 Horner terms (threads 0..63), in-place s2 update
    if (tid < 64) {
      int i = tid >> 3, j = tid & 7;
      float s2v = s2[tid];
      B4s[buf][tid] = s2v + (1.f / 3.f) * A4[i] * v[j];
      B3s[buf][tid] = s2v + 0.5f * A3[i] * v[j];
      s2[tid]       = s2v + A2[i] * v[j];
    }
#pragma unroll
    for (int c = 0; c < CH; ++c) s1[c] += v[c];
    __syncthreads();   // only barrier per step (double-buffered B4s/B3s)
    // Phase 3: level-3 Horner term (registers) + in-place s3 update
    float s3v0 = s3[p0], s3v1 = s3[p0 + 1];
    float bq4 = B4s[buf][q0], bq3 = B3s[buf][q0];
    float C0 = s3v0 + 0.5f * bq4 * v[j0];
    float C1 = s3v1 + 0.5f * bq4 * v[j0 + 1];
    s3[p0]     = s3v0 + bq3 * v[j0];
    s3[p0 + 1] = s3v1 + bq3 * v[j0 + 1];
    // Phase 4: rank-1 update of register-resident level-4 accumulator
#pragma unroll
    for (int l = 0; l < CH; ++l) {
      acc4[l]     += C0 * v[l];
      acc4[8 + l] += C1 * v[l];
    }
  }

  // Emit chunk signature: [s1 | s2 | s3 | s4]
  float* o = csig + (size_t)blockIdx.x * SIGD;
  if (tid < 8)  o[tid] = s1[tid];
  if (tid < 64) o[8 + tid] = s2[tid];
  o[72 + p0]     = s3[p0];
  o[72 + p0 + 1] = s3[p0 + 1];
#pragma unroll
  for (int r = 0; r < 2; ++r)
#pragma unroll
    for (int l = 0; l < CH; ++l)
      o[584 + (p0 + r) * CH + l] = acc4[r * CH + l];
}

// ---------------------------------------------------------------------------
// Kernel 2: fold chunk signatures with the Chen (tensor-algebra) product:
//   (S(x)T)_4 = a4 + a3(x)b1 + a2(x)b2 + a1(x)b3 + b4, etc.
// One workgroup per batch; a-levels live in regs/LDS, b-levels streamed in.
// Left-to-right combine == exact full-path signature.
// ---------------------------------------------------------------------------
__global__ __launch_bounds__(256) void chen_combine_kernel(const float* __restrict__ csig,
                                                           float* __restrict__ sig,
                                                           int nchunk)
{
  __shared__ float a2s[64], a3s[512];
  __shared__ float b2s[64], b3s[512];
  const int tid = threadIdx.x;
  const int p0  = 2 * tid;
  const float* base = csig + (size_t)blockIdx.x * nchunk * SIGD;

  float a1r[CH], a4r[16];
#pragma unroll
  for (int c = 0; c < CH; ++c) a1r[c] = base[c];
  if (tid < 64) a2s[tid] = base[8 + tid];
  a3s[p0]     = base[72 + p0];
  a3s[p0 + 1] = base[72 + p0 + 1];
#pragma unroll
  for (int r = 0; r < 16; ++r) a4r[r] = base[584 + p0 * CH + r];

  for (int c = 1; c < nchunk; ++c) {
    const float* bs = base + (size_t)c * SIGD;
    float b1r[CH];
#pragma unroll
    for (int k = 0; k < CH; ++k) b1r[k] = bs[k];
    if (tid < 64) b2s[tid] = bs[8 + tid];
    b3s[p0]     = bs[72 + p0];
    b3s[p0 + 1] = bs[72 + p0 + 1];
    const float* gb4 = bs + 584;
    __syncthreads();
    // n4 then n3 (both read OLD a1,a2,a3; a3 touched only at own rows)
#pragma unroll
    for (int r = 0; r < 2; ++r) {
      int p = p0 + r;
      int i = p >> 6, jk = p & 63, ij = p >> 3, k = p & 7;
      float a3v = a3s[p];
#pragma unroll
      for (int l = 0; l < CH; ++l)
        a4r[r * CH + l] += a3v * b1r[l] + a2s[ij] * b2s[(k << 3) | l]
                         + a1r[i] * b3s[(jk << 3) | l] + gb4[p * CH + l];
    }
#pragma unroll
    for (int r = 0; r < 2; ++r) {
      int p = p0 + r;
      a3s[p] = a3s[p] + a2s[p >> 3] * b1r[p & 7]
             + a1r[p >> 6] * b2s[p & 63] + b3s[p];
    }
    __syncthreads();   // all cross-thread reads of a2s done
    if (tid < 64) a2s[tid] = a2s[tid] + a1r[tid >> 3] * b1r[tid & 7] + b2s[tid];
#pragma unroll
    for (int k = 0; k < CH; ++k) a1r[k] += b1r[k];
    __syncthreads();   // b2s/b3s consumed before next chunk overwrites
  }

  float* o = sig + (size_t)blockIdx.x * SIGD;
  if (tid < 8)  o[tid] = a1r[tid];
  if (tid < 64) o[8 + tid] = a2s[tid];
  o[72 + p0]     = a3s[p0];
  o[72 + p0 + 1] = a3s[p0 + 1];
#pragma unroll
  for (int r = 0; r < 16; ++r) o[584 + p0 * CH + r] = a4r[r];
}

// ---------------------------------------------------------------------------
// Kernel 3: head GEMM out[128,10] = sig[128,4680] @ W^T + b via fp32 WMMA
// (V_WMMA_F32_16X16X4_F32, reference precision). One wave per 16-row tile.
// Cols n>=10 masked to zero so EXEC stays all-ones through the WMMA loop.
// ---------------------------------------------------------------------------
__global__ __launch_bounds__(32) void head_wmma_kernel(const float* __restrict__ sig,
                                                       const float* __restrict__ W,
                                                       const float* __restrict__ bias,
                                                       float* __restrict__ out)
{
  const int lane = threadIdx.x & 31;
  const int hh   = lane >> 4;          // half-wave: K sub-offset 0 or 2
  const int l15  = lane & 15;
  const int m    = blockIdx.x * 16 + l15;
  const float* arow = sig + (size_t)m * SIGD + 2 * hh;
  const int   wr    = (l15 < NOUT) ? l15 : 0;
  const float wmask = (l15 < NOUT) ? 1.f : 0.f;
  const float* brow = W + (size_t)wr * SIGD + 2 * hh;

  v8f acc = { 0.f, 0.f, 0.f, 0.f, 0.f, 0.f, 0.f, 0.f };
#pragma unroll 2
  for (int k0 = 0; k0 < SIGD; k0 += 4) {
    v2f a  = *(const v2f*)(arow + k0);
    v2f bb = *(const v2f*)(brow + k0);
    bb.x *= wmask;
    bb.y *= wmask;
    acc = __builtin_amdgcn_wmma_f32_16x16x4_f32(
        /*neg_a=*/false, a, /*neg_b=*/false, bb,
        /*c_mod=*/(short)0, acc, /*reuse_a=*/false, /*reuse_b=*/false);
  }

  float bv = (l15 < NOUT) ? bias[l15] : 0.f;
#pragma unroll
  for (int r = 0; r < 8; ++r) {
    int row = blockIdx.x * 16 + hh * 8 + r;   // D: M = r + 8*half, N = l15
    if (l15 < NOUT) out[row * NOUT + l15] = acc[r] + bv;
  }
}

extern "C" void kernel_launch(void* const* d_in, const int* in_sizes, int n_in,
                              void* d_out, int out_size, void* d_ws, size_t ws_size,
                              hipStream_t stream)
{
  (void)in_sizes; (void)n_in; (void)out_size;
  const float* X    = (const float*)d_in[0];   // (128, 512, 8) fp32
  const float* W    = (const float*)d_in[1];   // (10, 4680)  fp32
  const float* bias = (const float*)d_in[2];   // (10,)       fp32
  float* out = (float*)d_out;                  // (128, 10)   fp32

  // 4-way chunked scan (4x shorter critical path, 512 WGs) if scratch allows,
  // else single-chunk path. Layout: [chunk sigs | combined sigs].
  int nchunk = 4, clen = 128;
  size_t need = ((size_t)BATCH * nchunk + BATCH) * SIGD * sizeof(float);
  if (ws_size < need) { nchunk = 1; clen = LEN - 1; }

  float* csig = (float*)d_ws;                            // BATCH*nchunk x SIGD
  float* sig  = csig + (size_t)BATCH * nchunk * SIGD;    // BATCH x SIGD

  sig_chunk_kernel<<<BATCH * nchunk, 256, 0, stream>>>(X, csig, nchunk, clen);
  chen_combine_kernel<<<BATCH, 256, 0, stream>>>(csig, sig, nchunk);
  head_wmma_kernel<<<BATCH / 16, 32, 0, stream>>>(sig, W, bias, out);
}